// ScoringFunction_74749610819610
// MI455X (gfx1250) — compile-verified
//
#include <hip/hip_runtime.h>

typedef float v2f __attribute__((ext_vector_type(2)));
typedef float v8f __attribute__((ext_vector_type(8)));

#define NB    64    // batches
#define LQ    32    // query tokens
#define LD    512   // doc tokens
#define DD    128   // embedding dim
#define ND    16    // docs per query
#define NTOT  1024  // NB * ND
#define QSTR  132   // padded LDS row stride (floats): 132 % 64 == 4 -> conflict-free frags
#define DSTR  132
#define CHUNK 64    // doc tokens per LDS chunk
#define NCHUNK (LD / CHUNK)

__global__ __launch_bounds__(256)
void colbert_maxsim_kernel(const float* __restrict__ qemb,
                           const float* __restrict__ demb,
                           const unsigned char* __restrict__ qmask,
                           const unsigned char* __restrict__ dmask,
                           float* __restrict__ out) {
    __shared__ float q_lds[LQ * QSTR];        // 16.5 KB
    __shared__ float d_lds[CHUNK * DSTR];     // 33   KB
    __shared__ float wavemax[8][16];          // 0.5  KB

    const int n    = blockIdx.x;          // doc index 0..1023
    const int b    = n / ND;              // query index
    const int tid  = threadIdx.x;
    const int w    = tid >> 5;            // wave id 0..7
    const int lane = tid & 31;
    const int half = lane >> 4;           // 0: lanes 0-15, 1: lanes 16-31
    const int l16  = lane & 15;
    const int mt   = w & 1;               // M-tile (rows 0-15 or 16-31)
    const int nt   = w >> 1;              // N-tile within chunk (16 tokens)

    const float*         qbase = qemb + (size_t)b * LQ * DD;
    const float*         dbase = demb + (size_t)n * LD * DD;
    const unsigned char* qm    = qmask + (size_t)b * LQ;
    const unsigned char* dm    = dmask + (size_t)n * LD;

    // ---- stage query into LDS, query mask folded in (masked row -> zeros) ----
    // 32 rows * 32 float4 = 1024 float4, 4 per thread; each wave does whole rows.
    #pragma unroll
    for (int i = 0; i < 4; ++i) {
        int idx = tid + i * 256;
        int row = idx >> 5;               // 32 float4 per row
        int c4  = idx & 31;
        float4 v = make_float4(0.f, 0.f, 0.f, 0.f);
        if (qm[row])
            v = ((const float4*)(qbase + (size_t)row * DD))[c4];
        *(float4*)(&q_lds[row * QSTR + c4 * 4]) = v;
    }

    float runmax[8];
    #pragma unroll
    for (int j = 0; j < 8; ++j) runmax[j] = -__builtin_inff();

    for (int ch = 0; ch < NCHUNK; ++ch) {
        __syncthreads();                  // d_lds free to overwrite; q_lds visible
        const int tokBase = ch * CHUNK;

        // ---- stage doc chunk, doc mask folded in; masked rows NOT read from HBM ----
        // 64 rows * 32 float4 = 2048 float4, 8 per thread; branch is wave-uniform.
        #pragma unroll
        for (int i = 0; i < (CHUNK * (DD / 4)) / 256; ++i) {
            int idx = tid + i * 256;
            int row = idx >> 5;
            int c4  = idx & 31;
            float4 v = make_float4(0.f, 0.f, 0.f, 0.f);
            if (dm[tokBase + row])
                v = ((const float4*)(dbase + (size_t)(tokBase + row) * DD))[c4];
            *(float4*)(&d_lds[row * DSTR + c4 * 4]) = v;
        }
        // prefetch next chunk's live rows into cache (global_prefetch_b8)
        if (ch + 1 < NCHUNK) {
            int prow  = tid >> 2;             // 64 rows, 4 x 128B lines per row
            int pline = tid & 3;
            if (dm[tokBase + CHUNK + prow])
                __builtin_prefetch((const char*)(dbase + (size_t)(tokBase + CHUNK + prow) * DD)
                                       + pline * 128, 0, 1);
        }
        __syncthreads();

        // ---- 16x16 tile GEMM: K=128 via 32 x V_WMMA_F32_16X16X4_F32 ----
        v8f acc = {0.f, 0.f, 0.f, 0.f, 0.f, 0.f, 0.f, 0.f};
        const float* aptr = &q_lds[(mt * 16 + l16) * QSTR + 2 * half];
        const float* bptr = &d_lds[(nt * 16 + l16) * DSTR + 2 * half];
        #pragma unroll
        for (int k0 = 0; k0 < DD; k0 += 4) {
            v2f a  = *(const v2f*)(aptr + k0);   // {A[m][k], A[m][k+1]}, k = k0+2*half
            v2f bb = *(const v2f*)(bptr + k0);   // {d[tok][k], d[tok][k+1]}
            acc = __builtin_amdgcn_wmma_f32_16x16x4_f32(
                      false, a, false, bb, (short)0, acc, false, false);
        }
        // masked-out data is zero, so plain max == reference where(mask, sim, 0).max
        #pragma unroll
        for (int j = 0; j < 8; ++j) runmax[j] = fmaxf(runmax[j], acc[j]);
    }

    // ---- per-row max across the 16 columns held by this half-wave ----
    #pragma unroll
    for (int j = 0; j < 8; ++j) {
        float m = runmax[j];
        #pragma unroll
        for (int off = 1; off <= 8; off <<= 1)
            m = fmaxf(m, __shfl_xor(m, off, 32));
        runmax[j] = m;
    }
    if (l16 == 0) {
        // VGPR j holds row (j + 8*half) of this wave's M-tile
        #pragma unroll
        for (int j = 0; j < 8; ++j)
            wavemax[w][half * 8 + j] = runmax[j];
    }
    __syncthreads();

    // ---- combine waves (4 waves share each M-tile), sum 32 rows, store score ----
    if (tid < 32) {
        int rmt  = tid >> 4;              // which M-tile this row lives in
        int rloc = tid & 15;
        float m = wavemax[rmt][rloc];
        m = fmaxf(m, wavemax[rmt + 2][rloc]);
        m = fmaxf(m, wavemax[rmt + 4][rloc]);
        m = fmaxf(m, wavemax[rmt + 6][rloc]);
        #pragma unroll
        for (int off = 16; off >= 1; off >>= 1)
            m += __shfl_xor(m, off, 32);
        if (tid == 0) out[n] = m;
    }
}

extern "C" void kernel_launch(void* const* d_in, const int* in_sizes, int n_in,
                              void* d_out, int out_size, void* d_ws, size_t ws_size,
                              hipStream_t stream) {
    (void)in_sizes; (void)n_in; (void)d_ws; (void)ws_size; (void)out_size;
    const float*         q  = (const float*)d_in[0];
    const float*         d  = (const float*)d_in[1];
    const unsigned char* qm = (const unsigned char*)d_in[2];
    const unsigned char* dm = (const unsigned char*)d_in[3];
    float* out = (float*)d_out;
    hipLaunchKernelGGL(colbert_maxsim_kernel, dim3(NTOT), dim3(256), 0, stream,
                       q, d, qm, dm, out);
}